// HANModel_31344671326933
// MI455X (gfx1250) — compile-verified
//
#include <hip/hip_runtime.h>
#include <hip/hip_bf16.h>

#define HEADS 4
#define OUT_F 32
#define CH 128            // HEADS*OUT_F == CIN == 128 for both layers
#define NEG_SLOPE 0.2f
#define EPS 1e-16f
#define LDX 132           // padded LDS row stride (floats) to avoid bank conflicts

typedef float v2f __attribute__((ext_vector_type(2)));
typedef float v8f __attribute__((ext_vector_type(8)));

// ---------------- ordered-int float max helpers ----------------
__device__ __forceinline__ int f2ord(float f) {
    int i = __float_as_int(f);
    return (i >= 0) ? i : (i ^ 0x7FFFFFFF);
}
__device__ __forceinline__ float ord2f(int i) {
    return __int_as_float((i >= 0) ? i : (i ^ 0x7FFFFFFF));
}

// ---------------- GEMM: Y[M,128] = X[M,128] @ W[128,128] (fp32 WMMA) -------
// 128 threads = 4 waves per block; block covers 16 rows; each wave covers 32
// columns as two 16x16 tiles. A-fragments from LDS (ds_load_2addr_b64),
// B-fragments from global (W is 64KB -> WGP$/L2-resident).
__global__ __launch_bounds__(128)
void gat_gemm128(const float* __restrict__ X, const float* __restrict__ W,
                 float* __restrict__ Y) {
    __shared__ float xs[16 * LDX];
    const int m0  = blockIdx.x * 16;
    const int tid = threadIdx.x;

    // cooperative load of X[m0:m0+16, :] into LDS (float4, coalesced)
    const float4* Xv = (const float4*)(X + (size_t)m0 * CH);
    #pragma unroll
    for (int i = tid; i < 16 * (CH / 4); i += 128) {
        int row = i >> 5;           // 32 float4 per row
        int c4  = i & 31;
        float4 v = Xv[row * 32 + c4];
        float* d = &xs[row * LDX + c4 * 4];
        d[0] = v.x; d[1] = v.y; d[2] = v.z; d[3] = v.w;
    }
    __syncthreads();

    const int wave = tid >> 5;       // 0..3
    const int lane = tid & 31;
    const int half = lane >> 4;      // K lo/hi pair select (ISA A/B layout)
    const int l16  = lane & 15;
    const int n0   = wave * 32;

    v8f acc0 = {};
    v8f acc1 = {};
    #pragma unroll 4
    for (int k = 0; k < CH; k += 4) {
        const int kk = k + half * 2;
        v2f a, b0, b1;
        // A 16x4: lanes 0-15 -> K={k,k+1}, lanes 16-31 -> K={k+2,k+3}
        a.x = xs[l16 * LDX + kk];
        a.y = xs[l16 * LDX + kk + 1];
        // B 4x16: N on lanes, same K split
        const float* Wp = W + (size_t)kk * CH;
        b0.x = Wp[n0 + l16];
        b0.y = Wp[CH + n0 + l16];
        b1.x = Wp[n0 + 16 + l16];
        b1.y = Wp[CH + n0 + 16 + l16];
        acc0 = __builtin_amdgcn_wmma_f32_16x16x4_f32(false, a, false, b0,
                                                     (short)0, acc0, false, false);
        acc1 = __builtin_amdgcn_wmma_f32_16x16x4_f32(false, a, false, b1,
                                                     (short)0, acc1, false, false);
    }
    // C/D layout: VGPR r -> M = r (lanes 0-15) / r+8 (lanes 16-31), N = lane%16
    const int row_base = m0 + half * 8;
    #pragma unroll
    for (int r = 0; r < 8; ++r) {
        Y[(size_t)(row_base + r) * CH + n0 + l16]      = acc0[r];
        Y[(size_t)(row_base + r) * CH + n0 + 16 + l16] = acc1[r];
    }
}

// ------------- per-(node,head) attention dots + accumulator init -----------
__global__ void gat_node_alpha(const float* __restrict__ h,
                               const float* __restrict__ a_src,
                               const float* __restrict__ a_dst,
                               float* __restrict__ alpha_s,
                               float* __restrict__ alpha_d,
                               int*   __restrict__ m_ord,
                               float* __restrict__ s_sum, int NH) {
    int i = blockIdx.x * blockDim.x + threadIdx.x;
    if (i >= NH) return;
    const int hd = i & (HEADS - 1);
    const float* hp  = h + (size_t)i * OUT_F;     // h[n][head][f], i = n*H+head
    const float* asv = a_src + hd * OUT_F;
    const float* adv = a_dst + hd * OUT_F;
    float ss = 0.f, dd = 0.f;
    #pragma unroll
    for (int f = 0; f < OUT_F; ++f) {
        float v = hp[f];
        ss = fmaf(v, asv[f], ss);
        dd = fmaf(v, adv[f], dd);
    }
    alpha_s[i] = ss;
    alpha_d[i] = dd;
    m_ord[i]   = f2ord(-INFINITY);
    s_sum[i]   = 0.f;
}

// ------------- seed output with bias (scatter-adds accumulate onto it) -----
__global__ void gat_init_out(float* __restrict__ out,
                             const float* __restrict__ bias, int total) {
    int i = blockIdx.x * blockDim.x + threadIdx.x;
    if (i >= total) return;
    out[i] = bias[i & (CH - 1)];
}

// ------------- edge pass 1: leaky-relu score, segment max ------------------
__global__ void gat_edge_max(const int* __restrict__ src,
                             const int* __restrict__ dst,
                             const float* __restrict__ alpha_s,
                             const float* __restrict__ alpha_d,
                             int* __restrict__ m_ord, int E, int Etot4) {
    int i = blockIdx.x * blockDim.x + threadIdx.x;
    if (i >= Etot4) return;
    int e  = i >> 2;
    int hd = i & 3;
    int sn, dn;
    if (e < E) { sn = src[e]; dn = dst[e]; } else { sn = dn = e - E; }
    float ev = alpha_s[sn * HEADS + hd] + alpha_d[dn * HEADS + hd];
    ev = ev > 0.f ? ev : NEG_SLOPE * ev;
    atomicMax(&m_ord[dn * HEADS + hd], f2ord(ev));
}

// ------------- edge pass 2: ex = exp(e - m), segment sum, stash ex ---------
__global__ void gat_edge_sum(const int* __restrict__ src,
                             const int* __restrict__ dst,
                             const float* __restrict__ alpha_s,
                             const float* __restrict__ alpha_d,
                             const int* __restrict__ m_ord,
                             float* __restrict__ s_sum,
                             float* __restrict__ ex_buf, int E, int Etot4) {
    int i = blockIdx.x * blockDim.x + threadIdx.x;
    if (i >= Etot4) return;
    int e  = i >> 2;
    int hd = i & 3;
    int sn, dn;
    if (e < E) { sn = src[e]; dn = dst[e]; } else { sn = dn = e - E; }
    float ev = alpha_s[sn * HEADS + hd] + alpha_d[dn * HEADS + hd];
    ev = ev > 0.f ? ev : NEG_SLOPE * ev;
    float ex = expf(ev - ord2f(m_ord[dn * HEADS + hd]));
    ex_buf[i] = ex;
    atomicAdd(&s_sum[dn * HEADS + hd], ex);
}

// ------------- edge pass 2.5: normalize ex -> alpha in place ---------------
__global__ void gat_edge_alpha(const int* __restrict__ dst,
                               const float* __restrict__ s_sum,
                               float* __restrict__ ex_buf, int E, int Etot4) {
    int i = blockIdx.x * blockDim.x + threadIdx.x;
    if (i >= Etot4) return;
    int e  = i >> 2;
    int hd = i & 3;
    int dn = (e < E) ? dst[e] : (e - E);
    ex_buf[i] = ex_buf[i] / (s_sum[dn * HEADS + hd] + EPS);
}

// ------------- edge pass 3: gather h[src], scatter alpha*h into out --------
// 32 threads per edge (exactly one wave per edge -> src/dst loads are
// wave-uniform), float4 per thread; atomics land in L2 (out = 51MB << 192MB).
__global__ void gat_edge_aggr(const int* __restrict__ src,
                              const int* __restrict__ dst,
                              const float* __restrict__ alpha_buf,
                              const float* __restrict__ h,
                              float* __restrict__ out, int E, int Etot32) {
    int i = blockIdx.x * blockDim.x + threadIdx.x;
    if (i >= Etot32) return;
    int e  = i >> 5;
    int q  = i & 31;         // float4 chunk index; head = q/8
    int hd = q >> 3;
    int sn, dn;
    if (e < E) { sn = src[e]; dn = dst[e]; } else { sn = dn = e - E; }
    float alpha = alpha_buf[e * HEADS + hd];
    const float4 hv = *(const float4*)(h + (size_t)sn * CH + q * 4);
    float* op = out + (size_t)dn * CH + q * 4;
    atomicAdd(op + 0, alpha * hv.x);
    atomicAdd(op + 1, alpha * hv.y);
    atomicAdd(op + 2, alpha * hv.z);
    atomicAdd(op + 3, alpha * hv.w);
}

// ---------------------------------------------------------------------------
static void run_layer(const float* x_in, const float* W, const float* a_s,
                      const float* a_d, const float* bias, float* h,
                      float* alpha_s, float* alpha_d, int* m_ord, float* s_sum,
                      float* ex_buf, float* out, const int* src, const int* dst,
                      int N, int E, hipStream_t stream) {
    const int Etot   = E + N;        // + one self-loop per node
    const int NH     = N * HEADS;
    const int Etot4  = Etot * HEADS;
    const int Etot32 = Etot * 32;

    gat_gemm128<<<N / 16, 128, 0, stream>>>(x_in, W, h);
    gat_node_alpha<<<(NH + 255) / 256, 256, 0, stream>>>(h, a_s, a_d, alpha_s,
                                                         alpha_d, m_ord, s_sum, NH);
    gat_init_out<<<(N * CH + 255) / 256, 256, 0, stream>>>(out, bias, N * CH);
    gat_edge_max<<<(Etot4 + 255) / 256, 256, 0, stream>>>(src, dst, alpha_s,
                                                          alpha_d, m_ord, E, Etot4);
    gat_edge_sum<<<(Etot4 + 255) / 256, 256, 0, stream>>>(src, dst, alpha_s,
                                                          alpha_d, m_ord, s_sum,
                                                          ex_buf, E, Etot4);
    gat_edge_alpha<<<(Etot4 + 255) / 256, 256, 0, stream>>>(dst, s_sum, ex_buf,
                                                            E, Etot4);
    gat_edge_aggr<<<(Etot32 + 255) / 256, 256, 0, stream>>>(src, dst, ex_buf,
                                                            h, out, E, Etot32);
}

extern "C" void kernel_launch(void* const* d_in, const int* in_sizes, int n_in,
                              void* d_out, int out_size, void* d_ws, size_t ws_size,
                              hipStream_t stream) {
    const float* x       = (const float*)d_in[0];
    const int*   eidx    = (const int*)d_in[1];   // [2,E] int32 (JAX x64 off)
    const float* W1      = (const float*)d_in[2];
    const float* a_src1  = (const float*)d_in[3];
    const float* a_dst1  = (const float*)d_in[4];
    const float* b1      = (const float*)d_in[5];
    const float* W2      = (const float*)d_in[6];
    const float* a_src2  = (const float*)d_in[7];
    const float* a_dst2  = (const float*)d_in[8];
    const float* b2      = (const float*)d_in[9];
    float* out = (float*)d_out;

    const int N = in_sizes[0] / CH;
    const int E = in_sizes[1] / 2;
    const int* src = eidx;
    const int* dst = eidx + E;
    const int Etot = E + N;

    // workspace carve-up (fp32 unless noted); ~125 MB total, L2-resident
    float* h       = (float*)d_ws;                  // N*128
    float* out1    = h + (size_t)N * CH;            // N*128
    float* alpha_s = out1 + (size_t)N * CH;         // N*4
    float* alpha_d = alpha_s + (size_t)N * HEADS;   // N*4
    float* s_sum   = alpha_d + (size_t)N * HEADS;   // N*4
    int*   m_ord   = (int*)(s_sum + (size_t)N * HEADS);    // N*4
    float* ex_buf  = (float*)(m_ord + (size_t)N * HEADS);  // (E+N)*4

    // layer 1: x -> out1
    run_layer(x, W1, a_src1, a_dst1, b1, h, alpha_s, alpha_d, m_ord, s_sum,
              ex_buf, out1, src, dst, N, E, stream);
    // layer 2: out1 -> d_out (h buffer reused)
    run_layer(out1, W2, a_src2, a_dst2, b2, h, alpha_s, alpha_d, m_ord, s_sum,
              ex_buf, out, src, dst, N, E, stream);
    (void)Etot; (void)n_in; (void)out_size; (void)ws_size;
}